// TargetEncGlobalAttention_63934883168403
// MI455X (gfx1250) — compile-verified
//
#include <hip/hip_runtime.h>

typedef __attribute__((ext_vector_type(2))) float v2f;
typedef __attribute__((ext_vector_type(8))) float v8f;
typedef __attribute__((__vector_size__(4 * sizeof(int)))) int v4i;

typedef __attribute__((address_space(1))) void g_void;   // global
typedef __attribute__((address_space(3))) void l_void;   // LDS
typedef __attribute__((address_space(1))) v4i  g_v4i;
typedef __attribute__((address_space(3))) v4i  l_v4i;

#define B_   16
#define T_   1024
#define S_   1024
#define D_   1024
#define TM   32        // target rows per workgroup
#define HSTR 1028      // padded h_t row stride (floats): conflict-free WMMA A reads
#define BSTR 36        // staged-B row stride (floats): conflict-free WMMA B reads
#define KC   32        // k-chunk depth (floats)
#define NCH  (D_ / KC) // 32 chunks per 1024-deep reduction
#define NBUF 3         // LDS ring depth

// ---- coalesced async staging of one 256-row x 32-float fp32 tile into LDS ----
__device__ __forceinline__ void stage_chunk(const float* gbase, float* buf,
                                            int c, int tid)
{
    const int rsub = tid >> 3;          // 0..31
    const int colf = (tid & 7) * 4;     // 0,4,...,28  (16B granules)
    #pragma unroll
    for (int p = 0; p < 8; ++p) {
        const int row = p * 32 + rsub;  // 0..255
        const float* gp = gbase + (size_t)row * D_ + c * KC + colf;
        float*       lp = buf + row * BSTR + colf;
        __builtin_amdgcn_global_load_async_to_lds_b128(
            (g_v4i*)(g_void*)gp, (l_v4i*)(l_void*)lp, 0, 0);
    }
}

// ---- staged NT-GEMM block: C[32x256] += A[32x1024] * B[256x1024]^T ----------
// A fragments come from aBase (global src rows or LDS h_t rows, +k indexing),
// B rows stream global->LDS via the async ring, fragments read as ds_load_b64.
__device__ __forceinline__ void gemm_block(const float* aBase, const float* gB,
                                           float* stg0, v8f acc[4],
                                           int tid, int q, int lm, int koff)
{
    stage_chunk(gB, stg0, 0, tid);                       // prologue
    for (int c = 0; c < NCH; ++c) {
        if (c + 1 < NCH) {
            stage_chunk(gB, stg0 + ((c + 1) % NBUF) * (256 * BSTR), c + 1, tid);
            // newest 8 outstanding ops are chunk c+1's; <=8 => chunk c landed
            asm volatile("s_wait_asynccnt 0x8" ::: "memory");
        } else {
            asm volatile("s_wait_asynccnt 0x0" ::: "memory");
        }
        __syncthreads();   // chunk c visible to all; ring depth 3 protects writers

        const float* buf = stg0 + (c % NBUF) * (256 * BSTR);
        #pragma unroll
        for (int kk = 0; kk < KC; kk += 4) {
            v2f a = *(const v2f*)(aBase + c * KC + kk);  // shared by 4 subtiles
            #pragma unroll
            for (int j = 0; j < 4; ++j) {
                const int rloc = q * 64 + j * 16 + lm;   // B/N row within block
                v2f bb = *(const v2f*)(buf + rloc * BSTR + kk + koff);
                acc[j] = __builtin_amdgcn_wmma_f32_16x16x4_f32(
                    false, a, false, bb, (short)0, acc[j], false, false);
            }
        }
    }
}

__global__ __launch_bounds__(256)
void TargetEncGlobalAttention_kernel(const float* __restrict__ src,
                                     const float* __restrict__ mem,
                                     const int*   __restrict__ lens,
                                     const float* __restrict__ W,
                                     float*       __restrict__ out)
{
    // 128.5 KB h_t tile + 3 x 36 KB async staging ring = ~236.5 KB of 320 KB WGP LDS
    __shared__ __align__(16) float hT[TM * HSTR];
    __shared__ __align__(16) float stg[NBUF][256 * BSTR];

    const int tid  = threadIdx.x;
    const int lane = tid & 31;
    const int w    = tid >> 5;     // wave 0..7
    const int msub = w & 1;        // 16-row half of the 32-row tile
    const int q    = w >> 1;       // 0..3 column-stripe owner
    const int lm   = lane & 15;    // M(A)/N(B,C) index per CDNA5 16x16x4 f32 layout
    const int hi   = lane >> 4;    // K-half select
    const int koff = hi * 2;

    const int b  = blockIdx.x >> 5;            // 32 tiles per batch
    const int t0 = (blockIdx.x & 31) * TM;

    const float NEG_INF = -__builtin_inff();

    // ========== Phase 1: h_t = src_tile @ W^T -> LDS (4 e-blocks) ==========
    {
        const float* aRow = src + ((size_t)(b * T_ + t0 + msub * 16 + lm)) * D_ + koff;
        for (int eb = 0; eb < 4; ++eb) {
            v8f acc[4] = {};
            gemm_block(aRow, W + (size_t)(eb * 256) * D_, &stg[0][0],
                       acc, tid, q, lm, koff);
            #pragma unroll
            for (int j = 0; j < 4; ++j) {
                const int e0 = eb * 256 + q * 64 + j * 16;
                #pragma unroll
                for (int r = 0; r < 8; ++r) {
                    const int row = msub * 16 + hi * 8 + r;  // C/D: vgpr r -> row r / r+8
                    hT[row * HSTR + e0 + lm] = acc[j][r];
                }
            }
        }
    }
    __syncthreads();

    // ===== Phase 2: masked logits = h_t @ mem_bank^T -> global (4 s-blocks) =====
    {
        const int len = lens[b];
        const float* aLds = hT + (msub * 16 + lm) * HSTR + koff;
        for (int sb = 0; sb < 4; ++sb) {
            v8f acc[4] = {};
            gemm_block(aLds, mem + (size_t)(b * S_ + sb * 256) * D_, &stg[0][0],
                       acc, tid, q, lm, koff);
            #pragma unroll
            for (int j = 0; j < 4; ++j) {
                const int s = sb * 256 + q * 64 + j * 16 + lm;
                const bool valid = (s < len);
                #pragma unroll
                for (int r = 0; r < 8; ++r) {
                    const int row = msub * 16 + hi * 8 + r;
                    float v = valid ? acc[j][r] : NEG_INF;
                    out[((size_t)(b * T_ + t0 + row)) * S_ + s] = v;
                }
            }
        }
    }
    __threadfence();   // release logits to L2 so all waves of this WG can re-read
    __syncthreads();

    // ========== Phase 3: row-wise softmax (one wave per 4 rows) ==========
    {
        #pragma unroll
        for (int rr = 0; rr < 4; ++rr) {
            const int row = w * 4 + rr;                       // 0..31
            float* orow = out + ((size_t)(b * T_ + t0 + row)) * S_;

            float vals[S_ / 32];                              // 32 regs/lane row cache
            float m = NEG_INF;
            #pragma unroll
            for (int j = 0; j < S_ / 32; ++j) {
                vals[j] = orow[j * 32 + lane];
                m = fmaxf(m, vals[j]);
            }
            #pragma unroll
            for (int off = 16; off > 0; off >>= 1)
                m = fmaxf(m, __shfl_xor(m, off, 32));

            float l = 0.0f;
            #pragma unroll
            for (int j = 0; j < S_ / 32; ++j) {
                vals[j] = __expf(vals[j] - m);                // exp(-inf - m) == 0
                l += vals[j];
            }
            #pragma unroll
            for (int off = 16; off > 0; off >>= 1)
                l += __shfl_xor(l, off, 32);

            const float inv = 1.0f / l;
            #pragma unroll
            for (int j = 0; j < S_ / 32; ++j)
                orow[j * 32 + lane] = vals[j] * inv;
        }
    }
}

extern "C" void kernel_launch(void* const* d_in, const int* in_sizes, int n_in,
                              void* d_out, int out_size, void* d_ws, size_t ws_size,
                              hipStream_t stream) {
    (void)in_sizes; (void)n_in; (void)d_ws; (void)ws_size; (void)out_size;
    const float* src  = (const float*)d_in[0];   // [B, TGT, D]
    const float* mem  = (const float*)d_in[1];   // [B, SRC, D]
    const int*   lens = (const int*)d_in[2];     // [B]
    const float* W    = (const float*)d_in[3];   // [D, D]
    float*       out  = (float*)d_out;           // [B, TGT, SRC]

    dim3 grid(B_ * (T_ / TM));   // 512 workgroups
    dim3 block(256);             // 8 wave32
    TargetEncGlobalAttention_kernel<<<grid, block, 0, stream>>>(src, mem, lens, W, out);
}